// MagicGAT_7404523618682
// MI455X (gfx1250) — compile-verified
//
#include <hip/hip_runtime.h>
#include <math.h>

// ---------------- problem constants (match reference) ----------------
#define GAT_N     50000
#define GAT_E     800000
#define GAT_ET    (GAT_E + GAT_N)   // edges + self loops
#define NEG_SLOPE 0.2f
#define GAT_EPS   1e-16f

typedef __attribute__((ext_vector_type(2))) float v2f;
typedef __attribute__((ext_vector_type(8))) float v8f;

// ---------------------------------------------------------------------
// Dense GEMM  C[M,N] = A[M,K] * B[K,N]   (fp32, WMMA 16x16x4)
// One wave32 per 16x16 output tile. M,N multiples of 16; K multiple of 4.
// A 16x4 frag: lanes 0-15 row=lane, K={0,1}; lanes 16-31 row=lane-16, K={2,3}
// B 4x16 frag: lanes 0-15 col=lane, K={0,1}; lanes 16-31 col=lane-16, K={2,3}
// C/D v8f:     VGPR r -> M = half*8 + r, N = lane&15
// ---------------------------------------------------------------------
__global__ void gat_gemm_wmma_f32(const float* __restrict__ A,
                                  const float* __restrict__ B,
                                  float* __restrict__ C,
                                  int M, int N, int K) {
  const int lane = threadIdx.x;                       // 0..31
  const int tile = blockIdx.x * blockDim.y + threadIdx.y;
  const int ntn  = N >> 4;
  const int ntm  = M >> 4;
  if (tile >= ntm * ntn) return;                      // uniform per wave
  const int tm   = tile / ntn;
  const int tn   = tile % ntn;
  const int half = lane >> 4;                         // 0 or 1
  const int idx  = lane & 15;

  const float* __restrict__ Arow = A + (size_t)(tm * 16 + idx) * K;
  const float* __restrict__ Bcol = B + (tn * 16 + idx);

  v8f acc = {0.f, 0.f, 0.f, 0.f, 0.f, 0.f, 0.f, 0.f};
  for (int k0 = 0; k0 < K; k0 += 4) {
    const int k = k0 + half * 2;
    v2f a;
    a.x = Arow[k];
    a.y = Arow[k + 1];
    v2f b;
    b.x = Bcol[(size_t)k * N];
    b.y = Bcol[(size_t)(k + 1) * N];
    acc = __builtin_amdgcn_wmma_f32_16x16x4_f32(
        /*neg_a=*/false, a, /*neg_b=*/false, b,
        /*c_mod=*/(short)0, acc, /*reuse_a=*/false, /*reuse_b=*/false);
  }

  float* __restrict__ Crow = C + (size_t)(tm * 16 + half * 8) * N + tn * 16 + idx;
#pragma unroll
  for (int r = 0; r < 8; ++r) Crow[(size_t)r * N] = acc[r];
}

// ---------------------------------------------------------------------
// alpha_s[n,h] = sum_c hlin[n,h*C+c]*a_src[h*C+c]; same for alpha_d
// ---------------------------------------------------------------------
__global__ void gat_alpha_kernel(const float* __restrict__ hlin,
                                 const float* __restrict__ a_src,
                                 const float* __restrict__ a_dst,
                                 float* __restrict__ as_out,
                                 float* __restrict__ ad_out,
                                 int H, int C) {
  const int i = blockIdx.x * blockDim.x + threadIdx.x;
  if (i >= GAT_N * H) return;
  const int n  = i / H;
  const int hd = i % H;
  const float* __restrict__ hp = hlin + (size_t)n * H * C + (size_t)hd * C;
  const float* __restrict__ sp = a_src + (size_t)hd * C;
  const float* __restrict__ dp = a_dst + (size_t)hd * C;
  float s = 0.f, d = 0.f;
  for (int c = 0; c < C; ++c) {
    const float v = hp[c];
    s = fmaf(v, sp[c], s);
    d = fmaf(v, dp[c], d);
  }
  as_out[i] = s;
  ad_out[i] = d;
}

// ---------------------------------------------------------------------
// init: emax=-inf, denom=0, out[n,k]=bias[k]   (launched over N*D threads)
// ---------------------------------------------------------------------
__global__ void gat_init_kernel(float* __restrict__ emax,
                                float* __restrict__ denom,
                                float* __restrict__ out,
                                const float* __restrict__ bias,
                                int H, int D) {
  const int i = blockIdx.x * blockDim.x + threadIdx.x;
  if (i < GAT_N * H) {
    emax[i]  = -INFINITY;
    denom[i] = 0.f;
  }
  if (i < GAT_N * D) {
    out[i] = bias[i % D];
  }
}

// float atomic max via signed-max / unsigned-min bit trick
__device__ __forceinline__ void atomicMaxF32(float* addr, float val) {
  if (val >= 0.f)
    atomicMax((int*)addr, __float_as_int(val));
  else
    atomicMin((unsigned int*)addr, __float_as_uint(val));
}

// ---------------------------------------------------------------------
// edge pass 1: e = leaky_relu(as[src]+ad[dst]); atomic segment max on dst
// ---------------------------------------------------------------------
__global__ void gat_edge_max_kernel(const long long* __restrict__ ei,
                                    const float* __restrict__ as_in,
                                    const float* __restrict__ ad_in,
                                    float* __restrict__ ebuf,
                                    float* __restrict__ emax,
                                    int H) {
  const long long i = (long long)blockIdx.x * blockDim.x + threadIdx.x;
  if (i >= (long long)GAT_ET * H) return;
  const int e  = (int)(i / H);
  const int hd = (int)(i % H);
  int s, d;
  if (e < GAT_E) {
    s = (int)ei[e];
    d = (int)ei[GAT_E + e];
  } else {
    s = d = e - GAT_E;
  }
  float v = as_in[(size_t)s * H + hd] + ad_in[(size_t)d * H + hd];
  v = (v > 0.f) ? v : NEG_SLOPE * v;
  ebuf[i] = v;
  atomicMaxF32(&emax[(size_t)d * H + hd], v);
}

// ---------------------------------------------------------------------
// edge pass 2: ex = exp(e - emax[dst]); atomic segment sum; store ex
// ---------------------------------------------------------------------
__global__ void gat_edge_exp_kernel(const long long* __restrict__ ei,
                                    float* __restrict__ ebuf,
                                    const float* __restrict__ emax,
                                    float* __restrict__ denom,
                                    int H) {
  const long long i = (long long)blockIdx.x * blockDim.x + threadIdx.x;
  if (i >= (long long)GAT_ET * H) return;
  const int e  = (int)(i / H);
  const int hd = (int)(i % H);
  int d;
  if (e < GAT_E) d = (int)ei[GAT_E + e];
  else           d = e - GAT_E;
  const float ex = expf(ebuf[i] - emax[(size_t)d * H + hd]);
  ebuf[i] = ex;
  atomicAdd(&denom[(size_t)d * H + hd], ex);
}

// ---------------------------------------------------------------------
// edge pass 3: out[dst, t] += hlin[src, t] * ex/(denom[dst]+eps)
// one block (D threads) per edge -> coalesced gather + atomic scatter
// ---------------------------------------------------------------------
__global__ void gat_edge_scatter_kernel(const long long* __restrict__ ei,
                                        const float* __restrict__ hlin,
                                        const float* __restrict__ ebuf,
                                        const float* __restrict__ denom,
                                        float* __restrict__ out,
                                        int H, int C) {
  const int e = blockIdx.x;
  const int t = threadIdx.x;          // 0..H*C-1
  int s, d;
  if (e < GAT_E) {
    s = (int)ei[e];
    d = (int)ei[GAT_E + e];
  } else {
    s = d = e - GAT_E;
  }
  const int D  = H * C;
  const int hd = t / C;
  const float alpha = ebuf[(size_t)e * H + hd] /
                      (denom[(size_t)d * H + hd] + GAT_EPS);
  atomicAdd(&out[(size_t)d * D + t], hlin[(size_t)s * D + t] * alpha);
}

// ---------------------------------------------------------------------
// host-side: one full GAT layer
// ---------------------------------------------------------------------
static void run_gat_layer(const float* x_in, int in_dim,
                          const float* W, const float* a_src, const float* a_dst,
                          const float* bias, int H, int C,
                          const long long* ei,
                          float* out,     // [N, H*C]
                          float* hlin, float* asb, float* adb,
                          float* emax, float* denom, float* ebuf,
                          hipStream_t stream) {
  const int D = H * C;

  // 1) GEMM: hlin = x_in @ W
  {
    const int tiles = (GAT_N / 16) * (D / 16);
    dim3 blk(32, 4);
    dim3 grd((tiles + 3) / 4);
    gat_gemm_wmma_f32<<<grd, blk, 0, stream>>>(x_in, W, hlin, GAT_N, D, in_dim);
  }
  // 2) per-(node,head) attention dot products
  {
    const int tot = GAT_N * H;
    gat_alpha_kernel<<<(tot + 255) / 256, 256, 0, stream>>>(hlin, a_src, a_dst,
                                                            asb, adb, H, C);
  }
  // 3) init emax/denom/out(bias)
  {
    const int tot = GAT_N * D;
    gat_init_kernel<<<(tot + 255) / 256, 256, 0, stream>>>(emax, denom, out,
                                                           bias, H, D);
  }
  // 4) edge segment max
  {
    const long long tot = (long long)GAT_ET * H;
    gat_edge_max_kernel<<<(unsigned)((tot + 255) / 256), 256, 0, stream>>>(
        ei, asb, adb, ebuf, emax, H);
  }
  // 5) edge exp + segment sum
  {
    const long long tot = (long long)GAT_ET * H;
    gat_edge_exp_kernel<<<(unsigned)((tot + 255) / 256), 256, 0, stream>>>(
        ei, ebuf, emax, denom, H);
  }
  // 6) weighted scatter-add
  {
    gat_edge_scatter_kernel<<<GAT_ET, D, 0, stream>>>(ei, hlin, ebuf, denom,
                                                      out, H, C);
  }
}

extern "C" void kernel_launch(void* const* d_in, const int* in_sizes, int n_in,
                              void* d_out, int out_size, void* d_ws, size_t ws_size,
                              hipStream_t stream) {
  const float*     x   = (const float*)d_in[0];
  const long long* ei  = (const long long*)d_in[1];  // int64 edge_index [2, E]
  const float*     W0  = (const float*)d_in[2];
  const float*     as0 = (const float*)d_in[3];
  const float*     ad0 = (const float*)d_in[4];
  const float*     b0  = (const float*)d_in[5];
  const float*     W1  = (const float*)d_in[6];
  const float*     as1 = (const float*)d_in[7];
  const float*     ad1 = (const float*)d_in[8];
  const float*     b1  = (const float*)d_in[9];
  const float*     W2  = (const float*)d_in[10];
  const float*     as2 = (const float*)d_in[11];
  const float*     ad2 = (const float*)d_in[12];
  const float*     b2  = (const float*)d_in[13];

  float* ws = (float*)d_ws;
  size_t off = 0;
  float* bufA  = ws + off; off += (size_t)GAT_N * 128;   // layer0 output
  float* bufB  = ws + off; off += (size_t)GAT_N * 128;   // layer1 output
  float* hlin  = ws + off; off += (size_t)GAT_N * 128;   // per-layer GEMM out
  float* asb   = ws + off; off += (size_t)GAT_N * 4;
  float* adb   = ws + off; off += (size_t)GAT_N * 4;
  float* emax  = ws + off; off += (size_t)GAT_N * 4;
  float* denom = ws + off; off += (size_t)GAT_N * 4;
  float* ebuf  = ws + off; off += (size_t)GAT_ET * 4;
  (void)off; (void)ws_size; (void)in_sizes; (void)n_in; (void)out_size;

  // layer 0: 128 -> 4 heads x 32
  run_gat_layer(x,    128, W0, as0, ad0, b0, 4, 32, ei, bufA,
                hlin, asb, adb, emax, denom, ebuf, stream);
  // layer 1: 128 -> 4 heads x 32
  run_gat_layer(bufA, 128, W1, as1, ad1, b1, 4, 32, ei, bufB,
                hlin, asb, adb, emax, denom, ebuf, stream);
  // layer 2: 128 -> 1 head x 64, writes final output
  run_gat_layer(bufB, 128, W2, as2, ad2, b2, 1, 64, ei, (float*)d_out,
                hlin, asb, adb, emax, denom, ebuf, stream);
}